// LocContrastiveLoss_72636486910299
// MI455X (gfx1250) — compile-verified
//
#include <hip/hip_runtime.h>
#include <hip/hip_bf16.h>

#define B_    4
#define C_    64
#define H_    512
#define W_    512
#define HW_   (H_ * W_)
#define NBOX  30
#define TOPK_ 10

typedef float v2f __attribute__((ext_vector_type(2)));
typedef float v8f __attribute__((ext_vector_type(8)));

// ---------------------------------------------------------------------------
// Kernel 1: intensity[b,h,w] = sqrt(sum_c loc[b,c,h,w]^2)
// float4 per thread, coalesced B128 loads along w; channel loop strides H*W.
// ---------------------------------------------------------------------------
__global__ void intensity_kernel(const float* __restrict__ loc,
                                 float* __restrict__ inten) {
  int t = blockIdx.x * blockDim.x + threadIdx.x;   // one thread per 4 pixels
  int b = t / (HW_ / 4);
  int q = t - b * (HW_ / 4);
  const float4* p = reinterpret_cast<const float4*>(loc + (size_t)b * C_ * HW_) + q;
  float4 acc = make_float4(0.f, 0.f, 0.f, 0.f);
#pragma unroll 4
  for (int c = 0; c < C_; ++c) {
    float4 v = p[(size_t)c * (HW_ / 4)];
    acc.x += v.x * v.x; acc.y += v.y * v.y;
    acc.z += v.z * v.z; acc.w += v.w * v.w;
  }
  float4 o;
  o.x = sqrtf(acc.x); o.y = sqrtf(acc.y);
  o.z = sqrtf(acc.z); o.w = sqrtf(acc.w);
  reinterpret_cast<float4*>(inten)[t] = o;
}

// ---------------------------------------------------------------------------
// Kernel 2: per-batch 3x3 local-max peaks (intensity > 0.5), global top-10.
// One 256-thread block per batch. Private sorted top-10 per thread, then 10
// rounds of shared-memory argmax merge. Invalid slots get value -inf.
// ---------------------------------------------------------------------------
__global__ void peaks_kernel(const float* __restrict__ inten,
                             int* __restrict__ pidx,
                             float* __restrict__ pval) {
  const int b = blockIdx.x;
  const int tid = threadIdx.x;
  const float* I = inten + (size_t)b * HW_;
  const float NEG = -__builtin_inff();

  float tv[TOPK_]; int ti[TOPK_];
  for (int k = 0; k < TOPK_; ++k) { tv[k] = NEG; ti[k] = -1; }

  for (int i = tid; i < HW_; i += 256) {
    float v = I[i];
    if (!(v > 0.5f)) continue;
    int h = i >> 9, w = i & (W_ - 1);
    bool pk = true;
    for (int dy = -1; dy <= 1; ++dy)
      for (int dx = -1; dx <= 1; ++dx) {
        if (dy == 0 && dx == 0) continue;
        int nh = h + dy, nw = w + dx;
        if ((unsigned)nh < (unsigned)H_ && (unsigned)nw < (unsigned)W_)
          pk = pk && (v >= I[nh * W_ + nw]);
      }
    if (pk && v > tv[TOPK_ - 1]) {
      int p = TOPK_ - 1;
      while (p > 0 && tv[p - 1] < v) { tv[p] = tv[p - 1]; ti[p] = ti[p - 1]; --p; }
      tv[p] = v; ti[p] = i;
    }
  }

  __shared__ float rv[256];
  __shared__ int   ridx[256];
  __shared__ int   rtid[256];
  __shared__ int   winner;

  int p = 0;
  for (int r = 0; r < TOPK_; ++r) {
    float v = (p < TOPK_) ? tv[p] : NEG;
    int  ix = (p < TOPK_) ? ti[p] : -1;
    rv[tid] = v; ridx[tid] = ix; rtid[tid] = tid;
    __syncthreads();
    for (int s = 128; s > 0; s >>= 1) {
      if (tid < s && rv[tid + s] > rv[tid]) {
        rv[tid] = rv[tid + s]; ridx[tid] = ridx[tid + s]; rtid[tid] = rtid[tid + s];
      }
      __syncthreads();
    }
    if (tid == 0) {
      pval[b * TOPK_ + r] = rv[0];
      pidx[b * TOPK_ + r] = (ridx[0] < 0) ? 0 : ridx[0];
      winner = rtid[0];
    }
    __syncthreads();
    if (tid == winner) ++p;
    __syncthreads();
  }
}

// ---------------------------------------------------------------------------
// Kernel 3: gather+normalize rows, cosine sim via V_WMMA_F32_16X16X4_F32,
// masked relu mean, average over batch. 1 block, 4 waves (wave32) = 1/batch.
// ---------------------------------------------------------------------------
__global__ void __launch_bounds__(128) loss_kernel(
    const float* __restrict__ loc, const float* __restrict__ det,
    const float* __restrict__ boxes, const int* __restrict__ pidx,
    const float* __restrict__ pval, float* __restrict__ out) {
  __shared__ float dnS[B_][32][C_];   // normalized det rows (pad 30->32)
  __shared__ float pnS[B_][16][C_];   // normalized peak rows (pad 10->16)
  __shared__ int   staticF[B_][32];
  __shared__ int   validF[B_][16];
  __shared__ float lsumS[128];
  __shared__ float bl[B_];

  const int tid  = threadIdx.x;
  const int b    = tid >> 5;
  const int lane = tid & 31;

  // --- Phase A: det rows at gt box centers ---
  {
    int sflag = 0;
    if (lane < NBOX) {
      const float* bx = boxes + ((size_t)b * NBOX + lane) * 8;
      float x = bx[0], y = bx[1], cls = bx[7];
      float cxf = (x - (-59.9f)) / 119.8f * (float)W_;
      float cyf = (y - (-59.9f)) / 119.8f * (float)H_;
      int cx = (int)cxf; cx = cx < 0 ? 0 : (cx > W_ - 1 ? W_ - 1 : cx);
      int cy = (int)cyf; cy = cy < 0 ? 0 : (cy > H_ - 1 ? H_ - 1 : cy);
      const float* base = det + (size_t)b * C_ * HW_ + (size_t)cy * W_ + cx;
      float ss = 0.f;
      for (int c = 0; c < C_; ++c) {
        float v = base[(size_t)c * HW_];
        dnS[b][lane][c] = v; ss += v * v;
      }
      float sc = 1.0f / fmaxf(sqrtf(ss), 1e-8f);
      for (int c = 0; c < C_; ++c) dnS[b][lane][c] *= sc;
      sflag = (cls == 0.0f) ? 1 : 0;
    } else {
      for (int c = 0; c < C_; ++c) dnS[b][lane][c] = 0.f;
    }
    staticF[b][lane] = sflag;
  }

  // --- Phase B: peak rows ---
  if (lane < 16) {
    int vflag = 0;
    if (lane < TOPK_) {
      float v  = pval[b * TOPK_ + lane];
      int  idx = pidx[b * TOPK_ + lane];
      vflag = (v > 0.5f) ? 1 : 0;   // invalid slots are -inf
      if (vflag) {
        int py = idx >> 9, px = idx & (W_ - 1);
        const float* base = loc + (size_t)b * C_ * HW_ + (size_t)py * W_ + px;
        float ss = 0.f;
        for (int c = 0; c < C_; ++c) {
          float t2 = base[(size_t)c * HW_];
          pnS[b][lane][c] = t2; ss += t2 * t2;
        }
        float sc = 1.0f / fmaxf(sqrtf(ss), 1e-8f);
        for (int c = 0; c < C_; ++c) pnS[b][lane][c] *= sc;
      } else {
        for (int c = 0; c < C_; ++c) pnS[b][lane][c] = 0.f;
      }
    } else {
      for (int c = 0; c < C_; ++c) pnS[b][lane][c] = 0.f;
    }
    validF[b][lane] = vflag;
  }
  __syncthreads();

  // --- Phase C: sim = dn @ pn^T via WMMA f32 16x16x4, uniform control flow ---
  const int l16  = lane & 15;
  const int half = lane >> 4;
  float lsum = 0.f;
  for (int tile = 0; tile < 2; ++tile) {
    v8f acc = {0.f, 0.f, 0.f, 0.f, 0.f, 0.f, 0.f, 0.f};
    const int mb = tile * 16;
    for (int kk = 0; kk < 16; ++kk) {
      int k0 = kk * 4 + half * 2;
      // A 16x4: lane holds M=l16, K=k0..k0+1 ; B 4x16: lane holds K=k0..k0+1, N=l16
      v2f Av; Av.x = dnS[b][mb + l16][k0]; Av.y = dnS[b][mb + l16][k0 + 1];
      v2f Bv; Bv.x = pnS[b][l16][k0];      Bv.y = pnS[b][l16][k0 + 1];
      acc = __builtin_amdgcn_wmma_f32_16x16x4_f32(
          /*neg_a=*/false, Av, /*neg_b=*/false, Bv,
          /*c_mod=*/(short)0, acc, /*reuse_a=*/false, /*reuse_b=*/false);
    }
#pragma unroll
    for (int r = 0; r < 8; ++r) {
      int m = mb + r + half * 8;   // C/D layout: VGPR r -> M=r (+8 for hi lanes)
      int n = l16;
      if (m < NBOX && n < TOPK_ && staticF[b][m] && validF[b][n])
        lsum += fmaxf(acc[r] - 0.5f, 0.0f);
    }
  }
  lsumS[tid] = lsum;
  __syncthreads();

  if (lane == 0) {
    float s = 0.f; int ns = 0, nv = 0;
    for (int i = 0; i < 32; ++i) s += lsumS[b * 32 + i];
    for (int i = 0; i < NBOX;  ++i) ns += staticF[b][i];
    for (int i = 0; i < TOPK_; ++i) nv += validF[b][i];
    int np = ns * nv;
    bl[b] = (np > 0) ? (s / (float)np) : 0.0f;
  }
  __syncthreads();
  if (tid == 0) out[0] = (bl[0] + bl[1] + bl[2] + bl[3]) * 0.25f;
}

// ---------------------------------------------------------------------------
extern "C" void kernel_launch(void* const* d_in, const int* in_sizes, int n_in,
                              void* d_out, int out_size, void* d_ws, size_t ws_size,
                              hipStream_t stream) {
  const float* loc   = (const float*)d_in[0];
  const float* det   = (const float*)d_in[1];
  const float* boxes = (const float*)d_in[2];

  // workspace layout: intensity [B*H*W] f32 (4 MB), then peak idx/val
  float* inten = (float*)d_ws;
  int*   pidx  = (int*)((char*)d_ws + (size_t)B_ * HW_ * sizeof(float));
  float* pval  = (float*)(pidx + B_ * TOPK_);

  intensity_kernel<<<(B_ * HW_ / 4) / 256, 256, 0, stream>>>(loc, inten);
  peaks_kernel<<<B_, 256, 0, stream>>>(inten, pidx, pval);
  loss_kernel<<<1, 128, 0, stream>>>(loc, det, boxes, pidx, pval, (float*)d_out);
}